// DisentangledSelfAttention_84928683311130
// MI455X (gfx1250) — compile-verified
//
#include <hip/hip_runtime.h>

// ---------------------------------------------------------------------------
// DeBERTa disentangled self-attention, MI455X (gfx1250), all-f32 WMMA path.
// B=8, S=384, D=768, H=12, dh=64, MAX_POS=512 (P=1023 relative embeddings).
//
// Pipeline:
//   1) KpRel = rel_pos_emb @ Wpk, QpRel = rel_pos_emb @ Wpq   [1023 x 768]
//   2) q/k/v = x @ W* + biases, scattered to [B,H,S,64]
//   3) flash attention with banded c2p/p2c gathers (r = j - i + 511)
//   4) out = ctx @ Wo + bo
//
// WMMA fragments are single aligned 8-byte loads:
//   - A fragments: K-pairs contiguous along the row (natural layout)
//   - GEMM B fragments: W tile pair-swizzled in LDS ((k/2)*132 + 2n + (k&1))
//   - PV B fragments: V tile transposed in LDS (v_lds[d][j], pitch 36)
// GEMM global->LDS staging is register double-buffered so HBM/L2 latency
// overlaps the 64-WMMA compute phase of the previous K-chunk.
// ---------------------------------------------------------------------------

typedef float v2f __attribute__((ext_vector_type(2)));
typedef float v8f __attribute__((ext_vector_type(8)));

#define BN   8
#define SN   384
#define DN   768
#define HN   12
#define DH   64

static __device__ __forceinline__ v8f wmma_f32(v2f a, v2f b, v8f c) {
  // D = A(16x4 f32) x B(4x16 f32) + C(16x16 f32)
  return __builtin_amdgcn_wmma_f32_16x16x4_f32(
      /*neg_a=*/false, a, /*neg_b=*/false, b,
      /*c_mod=*/(short)0, c, /*reuse_a=*/false, /*reuse_b=*/false);
}

static __device__ __forceinline__ v8f v8f_zero() {
  v8f z;
#pragma unroll
  for (int i = 0; i < 8; ++i) z[i] = 0.0f;
  return z;
}

static __device__ __forceinline__ v2f ld2(const float* p) {
  return *(const v2f*)p;
}

// ---------------------------------------------------------------------------
// Generic GEMM: out = A[M,768] @ W[768,768] (+bias1 +bias2)
// scatter==0: out[m*768 + n]
// scatter==1: m=(b,s), n=(h,d) -> out[(((b*H+h)*S+s)*64 + d]
// 128 threads = 4 waves; 64x64 tile; wave w owns rows 16w..16w+15.
// Register double-buffered K-chunks of 64.
// ---------------------------------------------------------------------------
__global__ __launch_bounds__(128) void gemm_bias_kernel(
    const float* __restrict__ A, const float* __restrict__ W,
    const float* __restrict__ bias1, const float* __restrict__ bias2,
    float* __restrict__ out, int M, int scatter)
{
  __shared__ float a_lds[64 * 68];    // A tile, row-major, pitch 68
  __shared__ float w_lds[32 * 132];   // W tile, pair-swizzled: (k/2)*132 + 2n + (k&1)

  const int m0 = blockIdx.x * 64;
  const int n0 = blockIdx.y * 64;
  const int tid  = threadIdx.x;
  const int wv   = tid >> 5;
  const int lane = tid & 31;
  const int hi   = lane >> 4;   // half-wave select (K pair / M+8)
  const int lp   = lane & 15;

  // per-thread staging coordinates (8 float4 per tile)
  int ld_row[8], ld_col[8];
  const float* a_src[8];
#pragma unroll
  for (int r = 0; r < 8; ++r) {
    int idx = tid * 4 + r * 512;
    ld_row[r] = idx >> 6;
    ld_col[r] = idx & 63;
    int arow = m0 + ld_row[r]; if (arow >= M) arow = M - 1;  // clamped rows never stored
    a_src[r] = A + (size_t)arow * DN + ld_col[r];
  }

  v8f acc[4];
#pragma unroll
  for (int t = 0; t < 4; ++t) acc[t] = v8f_zero();

  // prologue: fetch K-chunk 0 into registers
  float4 a_reg[8], w_reg[8];
#pragma unroll
  for (int r = 0; r < 8; ++r) {
    a_reg[r] = *(const float4*)(a_src[r]);
    w_reg[r] = *(const float4*)(W + (size_t)ld_row[r] * DN + n0 + ld_col[r]);
  }

  for (int kc = 0; kc < DN; kc += 64) {
    __syncthreads();   // previous chunk's LDS consumers done
#pragma unroll
    for (int r = 0; r < 8; ++r) {
      *(float4*)(&a_lds[ld_row[r] * 68 + ld_col[r]]) = a_reg[r];
      int wb = (ld_row[r] >> 1) * 132 + (ld_row[r] & 1);
      w_lds[wb + 2 * (ld_col[r] + 0)] = w_reg[r].x;
      w_lds[wb + 2 * (ld_col[r] + 1)] = w_reg[r].y;
      w_lds[wb + 2 * (ld_col[r] + 2)] = w_reg[r].z;
      w_lds[wb + 2 * (ld_col[r] + 3)] = w_reg[r].w;
    }
    __syncthreads();

    // prefetch next K-chunk into registers; overlaps WMMA compute below
    if (kc + 64 < DN) {
#pragma unroll
      for (int r = 0; r < 8; ++r) {
        a_reg[r] = *(const float4*)(a_src[r] + kc + 64);
        w_reg[r] = *(const float4*)(W + (size_t)(kc + 64 + ld_row[r]) * DN + n0 + ld_col[r]);
      }
    }

#pragma unroll
    for (int kk = 0; kk < 16; ++kk) {
      v2f a = ld2(&a_lds[(16 * wv + lp) * 68 + 4 * kk + 2 * hi]);
      const int m = 2 * kk + hi;   // K-pair row in swizzled W
#pragma unroll
      for (int t = 0; t < 4; ++t) {
        v2f b = ld2(&w_lds[m * 132 + 2 * (16 * t + lp)]);
        acc[t] = wmma_f32(a, b, acc[t]);
      }
    }
  }

#pragma unroll
  for (int t = 0; t < 4; ++t) {
    const int n = n0 + 16 * t + lp;
    float bsum = (bias1 ? bias1[n] : 0.0f) + (bias2 ? bias2[n] : 0.0f);
#pragma unroll
    for (int g = 0; g < 8; ++g) {
      int row = m0 + 16 * wv + g + 8 * hi;
      if (row < M) {
        float val = acc[t][g] + bsum;
        if (scatter) {
          int b = row / SN, s = row - b * SN;
          int h = n >> 6,  d = n & 63;
          out[(((size_t)b * HN + h) * SN + s) * DH + d] = val;
        } else {
          out[(size_t)row * DN + n] = val;
        }
      }
    }
  }
}

// ---------------------------------------------------------------------------
// Flash attention with relative-position bands.
// Grid: (S/32, H, B); 64 threads = 2 waves; wave w owns query rows i0+16w..+15.
// Per 32-wide j tile: c2c (WMMA), QKp/KQp 64-wide windows (WMMA -> LDS),
// banded gather (idx = dj - di + 31), online softmax, P@V (WMMA).
// ---------------------------------------------------------------------------
__global__ __launch_bounds__(64) void attn_kernel(
    const float* __restrict__ q, const float* __restrict__ k,
    const float* __restrict__ v,
    const float* __restrict__ kpr,   // [1023][768]  rel_pos_emb @ Wpk
    const float* __restrict__ qpr,   // [1023][768]  rel_pos_emb @ Wpq
    float* __restrict__ ctx)         // [B*S][768]
{
  __shared__ float k_lds[32 * 68];   // k tile [32 j][64 d]; reused as P [32][34]
  __shared__ float v_lds[64 * 36];   // v tile TRANSPOSED [64 d][32 j], pitch 36
  __shared__ float qkp[32 * 68];     // QKp window [32 i][64 r]
  __shared__ float kqp[32 * 68];     // KQp window [32 j][64 r]

  const int i0 = blockIdx.x * 32;
  const int h  = blockIdx.y;
  const int b  = blockIdx.z;
  const int tid  = threadIdx.x;
  const int wv   = tid >> 5;
  const int lane = tid & 31;
  const int hi   = lane >> 4;
  const int lp   = lane & 15;

  // q strip A-fragments: rows i0+16w..+15, K = dh = 64 (16 WMMA k-steps)
  v2f qa[16];
  {
    const float* qrow =
        q + (((size_t)b * HN + h) * SN + (i0 + 16 * wv + lp)) * DH + 2 * hi;
#pragma unroll
    for (int kk = 0; kk < 16; ++kk) qa[kk] = ld2(qrow + 4 * kk);
  }

  v8f o[4];
#pragma unroll
  for (int t = 0; t < 4; ++t) o[t] = v8f_zero();
  float mrun[8], srun[8];
#pragma unroll
  for (int g = 0; g < 8; ++g) { mrun[g] = -3.0e38f; srun[g] = 0.0f; }

  for (int j0 = 0; j0 < SN; j0 += 32) {
    __syncthreads();   // previous iter's LDS consumers done
    {
      const float* kb = k + (((size_t)b * HN + h) * SN + j0) * DH;
      const float* vb = v + (((size_t)b * HN + h) * SN + j0) * DH;
#pragma unroll
      for (int r = 0; r < 8; ++r) {
        int idx = tid * 4 + r * 256;
        int row = idx >> 6, col = idx & 63;    // row = j in tile, col = d
        *(float4*)(&k_lds[row * 68 + col]) = *(const float4*)(kb + row * DH + col);
        float4 v4 = *(const float4*)(vb + row * DH + col);
        v_lds[(col + 0) * 36 + row] = v4.x;    // transpose: [d][j]
        v_lds[(col + 1) * 36 + row] = v4.y;
        v_lds[(col + 2) * 36 + row] = v4.z;
        v_lds[(col + 3) * 36 + row] = v4.w;
      }
    }
    __syncthreads();

    const int r0 = j0 - i0 + 480;   // r = j - i + 511; window idx = dj - di + 31

    // ---- c2c: q_strip[16x64] @ k[64(d) x 32(j)] ----
    v8f sc[2];
    sc[0] = v8f_zero(); sc[1] = v8f_zero();
#pragma unroll
    for (int kk = 0; kk < 16; ++kk) {
      v2f a = qa[kk];
#pragma unroll
      for (int t = 0; t < 2; ++t) {
        v2f bb = ld2(&k_lds[(16 * t + lp) * 68 + 4 * kk + 2 * hi]);
        sc[t] = wmma_f32(a, bb, sc[t]);
      }
    }

    // ---- QKp window: q_strip @ KpRel[r0..r0+63]^T  -> qkp LDS ----
#pragma unroll
    for (int t = 0; t < 4; ++t) {
      const float* kp = kpr + (size_t)(r0 + 16 * t + lp) * DN + h * DH + 2 * hi;
      v8f accq = v8f_zero();
#pragma unroll
      for (int kk = 0; kk < 16; ++kk) {
        accq = wmma_f32(qa[kk], ld2(kp + 4 * kk), accq);
      }
#pragma unroll
      for (int g = 0; g < 8; ++g)
        qkp[(16 * wv + g + 8 * hi) * 68 + 16 * t + lp] = accq[g];
    }

    // ---- KQp window: k_strip @ QpRel[r0..r0+63]^T  -> kqp LDS ----
#pragma unroll
    for (int t = 0; t < 4; ++t) {
      const float* qp = qpr + (size_t)(r0 + 16 * t + lp) * DN + h * DH + 2 * hi;
      v8f acck = v8f_zero();
#pragma unroll
      for (int kk = 0; kk < 16; ++kk) {
        v2f a = ld2(&k_lds[(16 * wv + lp) * 68 + 4 * kk + 2 * hi]);
        acck = wmma_f32(a, ld2(qp + 4 * kk), acck);
      }
#pragma unroll
      for (int g = 0; g < 8; ++g)
        kqp[(16 * wv + g + 8 * hi) * 68 + 16 * t + lp] = acck[g];
    }

    __syncthreads();   // qkp/kqp visible; k tile reads done -> reuse as P

    // ---- gather c2p/p2c, online softmax, write P (overlaying k_lds) ----
#pragma unroll
    for (int g = 0; g < 8; ++g) {
      const int dig = 16 * wv + g + 8 * hi;       // row in 32-row i block
      int i0x = lp - dig + 31;                    // band index for t=0
      float v0 = (sc[0][g] + qkp[dig * 68 + i0x] + kqp[lp * 68 + i0x]) * 0.125f;
      int i1x = i0x + 16;                         // band index for t=1
      float v1 = (sc[1][g] + qkp[dig * 68 + i1x] + kqp[(16 + lp) * 68 + i1x]) * 0.125f;

      float mx = fmaxf(v0, v1);
      mx = fmaxf(mx, __shfl_xor(mx, 1));
      mx = fmaxf(mx, __shfl_xor(mx, 2));
      mx = fmaxf(mx, __shfl_xor(mx, 4));
      mx = fmaxf(mx, __shfl_xor(mx, 8));
      float mn   = fmaxf(mrun[g], mx);
      float corr = __expf(mrun[g] - mn);
      mrun[g] = mn;
      float p0 = __expf(v0 - mn), p1 = __expf(v1 - mn);
      float ls = p0 + p1;
      ls += __shfl_xor(ls, 1);
      ls += __shfl_xor(ls, 2);
      ls += __shfl_xor(ls, 4);
      ls += __shfl_xor(ls, 8);
      srun[g] = srun[g] * corr + ls;
      o[0][g] *= corr; o[1][g] *= corr; o[2][g] *= corr; o[3][g] *= corr;
      k_lds[dig * 34 + lp]      = p0;   // P tile, pitch 34
      k_lds[dig * 34 + 16 + lp] = p1;
    }

    // ---- O += P[16x32] @ V[32x64] (V transposed in LDS -> b64 B-frags) ----
#pragma unroll
    for (int kk = 0; kk < 8; ++kk) {
      v2f a = ld2(&k_lds[(16 * wv + lp) * 34 + 4 * kk + 2 * hi]);
#pragma unroll
      for (int tn = 0; tn < 4; ++tn) {
        v2f bb = ld2(&v_lds[(16 * tn + lp) * 36 + 4 * kk + 2 * hi]);
        o[tn] = wmma_f32(a, bb, o[tn]);
      }
    }
  }

  // ---- normalize + store ctx[(b*S+s)][h*64+d] ----
#pragma unroll
  for (int tn = 0; tn < 4; ++tn) {
#pragma unroll
    for (int g = 0; g < 8; ++g) {
      int srow = i0 + 16 * wv + g + 8 * hi;
      int d    = 16 * tn + lp;
      ctx[(size_t)(b * SN + srow) * DN + h * DH + d] = o[tn][g] / srun[g];
    }
  }
}

// ---------------------------------------------------------------------------
extern "C" void kernel_launch(void* const* d_in, const int* in_sizes, int n_in,
                              void* d_out, int out_size, void* d_ws, size_t ws_size,
                              hipStream_t stream) {
  (void)in_sizes; (void)n_in; (void)out_size; (void)ws_size;
  const float* x   = (const float*)d_in[0];
  const float* rpe = (const float*)d_in[1];
  const float* Wq  = (const float*)d_in[2];
  const float* bq  = (const float*)d_in[3];
  const float* Wk  = (const float*)d_in[4];
  const float* bk  = (const float*)d_in[5];
  const float* Wv  = (const float*)d_in[6];
  const float* bv  = (const float*)d_in[7];
  const float* qb  = (const float*)d_in[8];
  const float* vb  = (const float*)d_in[9];
  const float* Wpk = (const float*)d_in[10];
  const float* Wpq = (const float*)d_in[11];
  const float* Wo  = (const float*)d_in[12];
  const float* bo  = (const float*)d_in[13];

  float* ws   = (float*)d_ws;
  const size_t QKV = (size_t)BN * HN * SN * DH;   // 2,359,296
  const size_t POS = (size_t)1023 * DN;           //   785,664
  float* qbuf = ws;
  float* kbuf = qbuf + QKV;
  float* vbuf = kbuf + QKV;
  float* kpr  = vbuf + QKV;
  float* qpr  = kpr + POS;
  float* ctx  = qpr + POS;

  dim3 blk(128);
  // relative-position projections (M = 1023)
  gemm_bias_kernel<<<dim3(16, 12), blk, 0, stream>>>(rpe, Wpk, nullptr, nullptr, kpr, 1023, 0);
  gemm_bias_kernel<<<dim3(16, 12), blk, 0, stream>>>(rpe, Wpq, nullptr, nullptr, qpr, 1023, 0);
  // q/k/v projections (M = B*S = 3072), scattered to [B,H,S,64]
  gemm_bias_kernel<<<dim3(48, 12), blk, 0, stream>>>(x, Wq, bq, qb,      qbuf, 3072, 1);
  gemm_bias_kernel<<<dim3(48, 12), blk, 0, stream>>>(x, Wk, bk, nullptr, kbuf, 3072, 1);
  gemm_bias_kernel<<<dim3(48, 12), blk, 0, stream>>>(x, Wv, bv, vb,      vbuf, 3072, 1);
  // fused disentangled flash attention
  attn_kernel<<<dim3(SN / 32, HN, BN), dim3(64), 0, stream>>>(qbuf, kbuf, vbuf, kpr, qpr, ctx);
  // output projection -> d_out
  gemm_bias_kernel<<<dim3(48, 12), blk, 0, stream>>>(ctx, Wo, bo, nullptr, (float*)d_out, 3072, 0);
}